// HiPPO_LTI_42477226557707
// MI455X (gfx1250) — compile-verified
//
#include <hip/hip_runtime.h>

typedef __attribute__((ext_vector_type(2))) float v2f;
typedef __attribute__((ext_vector_type(4))) float v4f;
typedef __attribute__((ext_vector_type(8))) float v8f;
typedef __attribute__((ext_vector_type(4))) unsigned int u32x4;
typedef __attribute__((ext_vector_type(8))) int i32x8;
typedef __attribute__((ext_vector_type(4))) int i32x4;

#define NS     256            // state dim
#define NB     64             // batch
#define LT     4096           // sequence length
#define NCHUNK 128
#define CLEN   32             // LT / NCHUNK
#define MATF   (NS * NS)      // 65536 floats per 256x256 matrix
#define STF    (NS * NB)      // 16384 floats per state block
#define LDSF   (MATF + STF)   // 81920 floats = 320 KB exactly

// Bank-conflict-free LDS layout for dA: per-row rotation by 4*(i&15) words.
// A-frag lanes (rows i0+lm, cols ka..ka+1) then hit banks ka+4*lm: distinct.
__device__ __forceinline__ int aidx(int i, int k) {
  return i * NS + ((k + ((i & 15) << 2)) & 255);
}
// State layout: k-adjacent pairs contiguous per b, with a 32-word rotation on
// odd row-pairs so the two EXEC halves read complementary bank sets.
__device__ __forceinline__ int sidx(int k, int b) {
  const int rp = k >> 1;
  return rp * 128 + (((b << 1) + ((rp & 1) << 5)) & 127) + (k & 1);
}

// ---- Tensor Data Mover descriptors (CDNA5 ISA ch.8): 1-D dword block copy --
__device__ __forceinline__ u32x4 tdm_g0(unsigned lds_off, const void* g) {
  unsigned long long ga = (unsigned long long)g;
  u32x4 d;
  d.x = 1u;                                            // count=1, user mode
  d.y = lds_off;                                       // lds_addr
  d.z = (unsigned)(ga & 0xffffffffu);                  // global_addr[31:0]
  d.w = (unsigned)((ga >> 32) & 0x01ffffffu) | (2u << 30);  // addr[56:32]|type=2
  return d;
}
__device__ __forceinline__ i32x8 tdm_g1_1d(int n) {    // n contiguous dwords
  i32x8 d = {};
  d[0] = (2 << 16);                  // data_size = 4B
  d[1] = (n & 0xffff) << 16;         // tensor_dim0[15:0]
  d[2] = ((n >> 16) & 0xffff) | (1 << 16);  // tensor_dim0[31:16], tensor_dim1=1
  d[3] = (n & 0xffff) << 16;         // tile_dim0 = n
  d[4] = 1;                          // tile_dim1 = 1
  d[5] = n;                          // tensor_dim0_stride = n
  d[6] = 0;
  d[7] = 0;
  return d;
}

// ---------------------------------------------------------------------------
// dst = src @ src (256x256 f32) via V_WMMA_F32_16X16X4_F32.
// 32 blocks x 256 thr (8 waves); one 16x16 tile per wave. L2-resident operand.
// ---------------------------------------------------------------------------
__global__ void k_square(const float* __restrict__ src, float* __restrict__ dst) {
  const int lane = threadIdx.x & 31;
  const int wave = threadIdx.x >> 5;
  const int half = lane >> 4;
  const int lm   = lane & 15;
  const int tile = blockIdx.x * 8 + wave;
  const int i0 = (tile >> 4) * 16;
  const int j0 = (tile & 15) * 16;
  v8f acc = {};
  for (int k0 = 0; k0 < NS; k0 += 4) {
    const int ka = k0 + 2 * half;
    v2f a, b;
    a.x = src[(i0 + lm) * NS + ka];
    a.y = src[(i0 + lm) * NS + ka + 1];
    b.x = src[ka * NS + j0 + lm];
    b.y = src[(ka + 1) * NS + j0 + lm];
    acc = __builtin_amdgcn_wmma_f32_16x16x4_f32(false, a, false, b,
                                                (short)0, acc, false, false);
  }
#pragma unroll
  for (int v = 0; v < 8; ++v)
    dst[(i0 + v + 8 * half) * NS + j0 + lm] = acc[v];
}

// ---------------------------------------------------------------------------
// Per-chunk sequential scan: S <- dA @ S + dB (x) x_t, CLEN steps.
// phase 0: S starts at 0; chunk-end state exported via TENSOR_STORE_FROM_LDS.
// phase 1: S starts at Tprev[chunk-1] via TENSOR_LOAD_TO_LDS (overlapped with
//          VALU-swizzled dA staging); every step stores out[t,b,i].
// 1 WG per chunk, 512 thr = 16 waves; wave w owns i-rows [16w,16w+16).
// ---------------------------------------------------------------------------
__global__ void k_chunk(const float* __restrict__ x,      // (LT, NB)
                        const float* __restrict__ dA,     // (NS, NS)
                        const float* __restrict__ dB,     // (NS)
                        const float* __restrict__ Tprev,  // scan result or null
                        float* __restrict__ Eout,         // phase0 chunk ends
                        float* __restrict__ out,          // phase1 (LT,NB,NS)
                        int phase) {
  extern __shared__ float smem[];
  float* dAs = smem;          // 65536 floats, rotated layout aidx()
  float* S   = smem + MATF;   // 16384 floats, paired layout sidx()
  const int tid  = threadIdx.x;
  const int lane = tid & 31, wave = tid >> 5;
  const int half = lane >> 4, lm = lane & 15;
  const int chunk = blockIdx.x;
  const int i0 = wave * 16;
  const bool tdm_init = (phase == 1) && (chunk > 0);
  const unsigned lds_off_S = (unsigned)(size_t)(void*)S;  // addr[31:0] = offset

#if __has_builtin(__builtin_amdgcn_tensor_load_to_lds)
  if (tdm_init && tid < 32) {   // wave 0 kicks the DMA before dA staging
    const i32x4 z4 = {};
    const i32x8 z8 = {};
    __builtin_amdgcn_tensor_load_to_lds(
        tdm_g0(lds_off_S, Tprev + (size_t)(chunk - 1) * STF),
        tdm_g1_1d(STF), z4, z4, z8, 0);
  }
#endif

  // stage dA into LDS with bank rotation (coalesced v4f reads)
  {
    const v4f* s4 = (const v4f*)dA;
    for (int r = 0; r < 32; ++r) {
      const int g = tid + r * 512;            // v4f group id, 16384 total
      const int i = g >> 6, k = (g & 63) << 2;
      *(v4f*)&dAs[aidx(i, k)] = s4[g];
    }
  }
  if (!tdm_init) {   // phase 0, or chunk 0 of phase 1: zero state
    v4f z = {};
    v4f* d4 = (v4f*)S;
    for (int r = 0; r < 8; ++r) d4[tid + r * 512] = z;
#if !__has_builtin(__builtin_amdgcn_tensor_load_to_lds)
  } else {           // fallback path if TDM builtin is unavailable
    const v4f* s4 = (const v4f*)(Tprev + (size_t)(chunk - 1) * STF);
    v4f* d4 = (v4f*)S;
    for (int r = 0; r < 8; ++r) d4[tid + r * 512] = s4[tid + r * 512];
#endif
  }
  float dbv[8];
#pragma unroll
  for (int v = 0; v < 8; ++v) dbv[v] = dB[i0 + v + 8 * half];
#if __has_builtin(__builtin_amdgcn_tensor_load_to_lds)
  if (tdm_init && tid < 32) __builtin_amdgcn_s_wait_tensorcnt(0);
#endif
  __syncthreads();

  const int t0 = chunk * CLEN;
  for (int k = 0; k < CLEN; ++k) {
    const int t = t0 + k;
    const float* xr = x + (size_t)t * NB;
    v8f acc[4];
#pragma unroll
    for (int bt = 0; bt < 4; ++bt) {
      const float xv = xr[bt * 16 + lm];
#pragma unroll
      for (int v = 0; v < 8; ++v) acc[bt][v] = dbv[v] * xv;   // u_t term as C
    }
    for (int k0 = 0; k0 < NS; k0 += 4) {
      const int ka = k0 + 2 * half;
      v2f a = *(const v2f*)&dAs[aidx(i0 + lm, ka)];   // conflict-free b64
#pragma unroll
      for (int bt = 0; bt < 4; ++bt) {
        v2f b = *(const v2f*)&S[sidx(ka, bt * 16 + lm)];  // conflict-free b64
        acc[bt] = __builtin_amdgcn_wmma_f32_16x16x4_f32(false, a, false, b,
                                                        (short)0, acc[bt],
                                                        false, false);
      }
    }
    __syncthreads();   // all reads of S done
#pragma unroll
    for (int bt = 0; bt < 4; ++bt)
#pragma unroll
      for (int v = 0; v < 8; v += 2) {
        v2f tpair = {acc[bt][v], acc[bt][v + 1]};
        *(v2f*)&S[sidx(i0 + v + 8 * half, bt * 16 + lm)] = tpair;
      }
    __syncthreads();   // S holds S_t
    if (phase == 1) {
      float* op = out + (size_t)t * (NB * NS);
#pragma unroll
      for (int bt = 0; bt < 4; ++bt) {
        float* p = op + (size_t)(bt * 16 + lm) * NS + i0 + 8 * half;
        v4f lo = {acc[bt][0], acc[bt][1], acc[bt][2], acc[bt][3]};
        v4f hi = {acc[bt][4], acc[bt][5], acc[bt][6], acc[bt][7]};
        *(v4f*)p = lo;
        *(v4f*)(p + 4) = hi;
      }
    }
  }
  if (phase == 0) {   // export chunk-end state (LDS -> global) via TDM
#if __has_builtin(__builtin_amdgcn_tensor_store_from_lds)
    if (tid < 32) {
      const i32x4 z4 = {};
      const i32x8 z8 = {};
      __builtin_amdgcn_tensor_store_from_lds(
          tdm_g0(lds_off_S, Eout + (size_t)chunk * STF),
          tdm_g1_1d(STF), z4, z4, z8, 0);
      __builtin_amdgcn_s_wait_tensorcnt(0);
    }
#else
    const v4f* s4 = (const v4f*)S;
    v4f* d4 = (v4f*)(Eout + (size_t)chunk * STF);
    for (int r = 0; r < 8; ++r) d4[tid + r * 512] = s4[tid + r * 512];
#endif
  }
}

// ---------------------------------------------------------------------------
// One Hillis-Steele level: Eout[j] = Ein[j] + M @ Ein[j-stride] (j >= stride).
// E blocks use the same sidx() pairing, so operand pairs are b64 in global.
// ---------------------------------------------------------------------------
__global__ void k_scanlevel(const float* __restrict__ Ein,
                            float* __restrict__ Eout,
                            const float* __restrict__ M, int stride) {
  const int j = blockIdx.x;
  const int tid = threadIdx.x;
  const float* src = Ein + (size_t)j * STF;
  float* dst = Eout + (size_t)j * STF;
  if (j < stride) {   // uniform branch: EXEC stays full
    const v4f* s4 = (const v4f*)src;
    v4f* d4 = (v4f*)dst;
    for (int r = 0; r < 8; ++r) d4[tid + r * 512] = s4[tid + r * 512];
    return;
  }
  const int lane = tid & 31, wave = tid >> 5;
  const int half = lane >> 4, lm = lane & 15;
  const int i0 = wave * 16;
  const float* prev = Ein + (size_t)(j - stride) * STF;
  v8f acc[4];
#pragma unroll
  for (int bt = 0; bt < 4; ++bt)
#pragma unroll
    for (int v = 0; v < 8; v += 2) {
      v2f tpair = *(const v2f*)&src[sidx(i0 + v + 8 * half, bt * 16 + lm)];
      acc[bt][v] = tpair.x;
      acc[bt][v + 1] = tpair.y;
    }
  for (int k0 = 0; k0 < NS; k0 += 4) {
    const int ka = k0 + 2 * half;
    v2f a = *(const v2f*)&M[(i0 + lm) * NS + ka];
#pragma unroll
    for (int bt = 0; bt < 4; ++bt) {
      v2f b = *(const v2f*)&prev[sidx(ka, bt * 16 + lm)];
      acc[bt] = __builtin_amdgcn_wmma_f32_16x16x4_f32(false, a, false, b,
                                                      (short)0, acc[bt],
                                                      false, false);
    }
  }
#pragma unroll
  for (int bt = 0; bt < 4; ++bt)
#pragma unroll
    for (int v = 0; v < 8; v += 2) {
      v2f tpair = {acc[bt][v], acc[bt][v + 1]};
      *(v2f*)&dst[sidx(i0 + v + 8 * half, bt * 16 + lm)] = tpair;
    }
}

// ---------------------------------------------------------------------------
extern "C" void kernel_launch(void* const* d_in, const int* in_sizes, int n_in,
                              void* d_out, int out_size, void* d_ws,
                              size_t ws_size, hipStream_t stream) {
  (void)in_sizes; (void)n_in; (void)out_size; (void)ws_size;
  const float* x  = (const float*)d_in[0];   // (4096, 64)
  const float* dA = (const float*)d_in[1];   // (256, 256)
  const float* dB = (const float*)d_in[2];   // (256)
  float* out = (float*)d_out;                // (4096, 64, 256)

  float* Q  = (float*)d_ws;                  // Q_k = dA^(2^k), k=1..11
  float* EA = Q + 11 * MATF;                 // 128 chunk states
  float* EB = EA + NCHUNK * STF;             // ping-pong buffer

  (void)hipFuncSetAttribute(reinterpret_cast<const void*>(k_chunk),
                            hipFuncAttributeMaxDynamicSharedMemorySize,
                            LDSF * 4);

  // power chain: Q1 = dA^2, ..., Q11 = dA^2048
  k_square<<<32, 256, 0, stream>>>(dA, Q);
  for (int k = 1; k < 11; ++k)
    k_square<<<32, 256, 0, stream>>>(Q + (size_t)(k - 1) * MATF,
                                     Q + (size_t)k * MATF);

  // phase 1: local chunk scans from zero -> chunk-end states in EA
  k_chunk<<<NCHUNK, 512, LDSF * 4, stream>>>(x, dA, dB, nullptr, EA, nullptr, 0);

  // phase 2: 7-level scan; level s uses M = dA^(32*2^s) = Q_(5+s)
  float* bufs[2] = {EA, EB};
  int cur = 0;
  for (int s = 0; s < 7; ++s) {
    k_scanlevel<<<NCHUNK, 512, 0, stream>>>(bufs[cur], bufs[cur ^ 1],
                                            Q + (size_t)(4 + s) * MATF, 1 << s);
    cur ^= 1;
  }

  // phase 3: re-run chunks from true incoming states, emit all outputs
  k_chunk<<<NCHUNK, 512, LDSF * 4, stream>>>(x, dA, dB, bufs[cur], nullptr,
                                             out, 1);
}